// M_41077067219161
// MI455X (gfx1250) — compile-verified
//
#include <hip/hip_runtime.h>

// Problem dims (fixed by the reference)
#define B_ 8
#define R_ 128
#define H_ 4096
#define W_ 4096

// k_x LDS staging geometry
#define KC_     128                 // K-slice width (floats)
#define LROW_   132                 // LDS row stride: 132 % 64 == 4 -> conflict-free
#define SLICE_F (R_ * LROW_)        // floats per slice buffer (16896)
#define NSLICE  (W_ / KC_)          // 32

typedef __attribute__((ext_vector_type(2))) float v2f;
typedef __attribute__((ext_vector_type(8))) float v8f;

#define HAVE_TDM __has_builtin(__builtin_amdgcn_tensor_load_to_lds)

// V_WMMA_F32_16X16X4_F32: D(16x16,f32) = A(16x4,f32) * B(4x16,f32) + C
__device__ __forceinline__ v8f wmma_f32(v2f a, v2f b, v8f c) {
    return __builtin_amdgcn_wmma_f32_16x16x4_f32(false, a, false, b,
                                                 (short)0, c, false, false);
}

#if HAVE_TDM
// Issue one TDM load of a [R_ x KC_] f32 tile (row stride W_ elements in
// memory) into LDS at byte offset lds_off, padding 4 DWORDs after every 128
// DWORDs so the LDS row stride becomes LROW_ = 132 floats.
// D# layout per CDNA5 ISA 8.3-8.5.
__device__ __forceinline__ void tdm_load_slice(const float* gsrc, unsigned lds_off) {
    typedef __attribute__((ext_vector_type(4))) unsigned u32x4;
    typedef __attribute__((ext_vector_type(8))) int      i32x8;
    typedef __attribute__((ext_vector_type(4))) int      i32x4;

    unsigned long long ga = (unsigned long long)(size_t)gsrc;
    u32x4 g0;
    g0.x = 1u;                                       // count=1, user mode
    g0.y = lds_off;                                  // lds_addr (bytes)
    g0.z = (unsigned)(ga & 0xFFFFFFFFu);             // global_addr[31:0]
    g0.w = (unsigned)((ga >> 32) & 0x01FFFFFFu)      // global_addr[56:32]
         | (2u << 30);                               // type = 2 ("image")

    i32x8 g1;
    g1[0] = (2 << 16)                                // data_size = 4 bytes
          | (1 << 20)                                // pad_enable
          | (6 << 22)                                // pad_interval: 128 DWORDs
          | (3 << 25);                               // pad_amount: 4 DWORDs
    g1[1] = (int)((W_ & 0xFFFF) << 16);              // tensor_dim0[15:0]
    g1[2] = (int)(((W_ >> 16) & 0xFFFF)              // tensor_dim0[31:16]
          | ((R_ & 0xFFFF) << 16));                  // tensor_dim1[15:0]
    g1[3] = (int)(((R_ >> 16) & 0xFFFF)              // tensor_dim1[31:16]
          | (KC_ << 16));                            // tile_dim0 = 128
    g1[4] = (int)R_;                                 // tile_dim1 = 128, tile_dim2 = 0
    g1[5] = (int)W_;                                 // tensor_dim0_stride[31:0] = 4096
    g1[6] = 0;                                       // stride hi / dim1_stride lo
    g1[7] = 0;

    i32x4 zz = {0, 0, 0, 0};
#if __clang_major__ >= 23
    i32x8 z8 = {0, 0, 0, 0, 0, 0, 0, 0};
    __builtin_amdgcn_tensor_load_to_lds(g0, g1, zz, zz, z8, 0);
#else
    __builtin_amdgcn_tensor_load_to_lds(g0, g1, zz, zz, 0);
#endif
}
#endif // HAVE_TDM

// ---------------------------------------------------------------------------
// Kernel 1: G = N * N^T per batch. One wave computes a 16x128 strip (8 tiles).
// ---------------------------------------------------------------------------
__global__ __launch_bounds__(32) void k_gram(const float* __restrict__ Nmat,
                                             float* __restrict__ G) {
    const int mt   = blockIdx.x;
    const int b    = blockIdx.y;
    const int lane = threadIdx.x;
    const int half = lane >> 4;
    const int lm   = lane & 15;

    const float* Nb = Nmat + (size_t)b * R_ * W_;
    const float* arow = Nb + (size_t)(mt * 16 + lm) * W_;

    v8f z = {0.f, 0.f, 0.f, 0.f, 0.f, 0.f, 0.f, 0.f};
    v8f acc[8];
#pragma unroll
    for (int j = 0; j < 8; ++j) acc[j] = z;

    for (int k = 0; k < W_; k += 4) {
        const int kk = k + 2 * half;
        v2f a = *(const v2f*)(arow + kk);
#pragma unroll
        for (int j = 0; j < 8; ++j) {
            v2f bb = *(const v2f*)(Nb + (size_t)(j * 16 + lm) * W_ + kk);
            acc[j] = wmma_f32(a, bb, acc[j]);
        }
    }

    float* Gb = G + (size_t)b * R_ * R_;
#pragma unroll
    for (int j = 0; j < 8; ++j)
#pragma unroll
        for (int g = 0; g < 8; ++g) {
            const int row = mt * 16 + g + 8 * half;
            const int col = j * 16 + lm;
            Gb[row * R_ + col] = acc[j][g];
        }
}

// ---------------------------------------------------------------------------
// Kernel 2: per batch, in-place Gauss-Jordan inversion of C = alpha*I + 2g*G
// (SPD -> pinv == inv). Output pre-scaled by 2*gamma. LDS stride 129.
// ---------------------------------------------------------------------------
__global__ __launch_bounds__(256) void k_inv(const float* __restrict__ G,
                                             const float* __restrict__ alpha,
                                             const float* __restrict__ gamma,
                                             float* __restrict__ Cinv) {
    extern __shared__ float sm[];
#define MS(i, j) sm[(i) * 129 + (j)]
    const int b   = blockIdx.x;
    const int tid = threadIdx.x;
    const int row = tid & 127;
    const int j0  = (tid >> 7) * 64;

    const float a  = alpha[b];
    const float g2 = 2.f * gamma[b];
    const float* Gb = G + (size_t)b * R_ * R_;

    for (int j = j0; j < j0 + 64; ++j)
        MS(row, j) = g2 * Gb[row * R_ + j] + (row == j ? a : 0.f);
    __syncthreads();

    for (int k = 0; k < R_; ++k) {
        const float p = 1.0f / MS(k, k);
        const float f = MS(row, k);
        __syncthreads();
        if (row == k) {
            for (int j = j0; j < j0 + 64; ++j)
                MS(k, j) = (j == k) ? p : MS(k, j) * p;
        }
        __syncthreads();
        if (row != k) {
            for (int j = j0; j < j0 + 64; ++j) {
                const float rkj = MS(k, j);
                MS(row, j) = (j == k) ? (-f * p) : fmaf(-f, rkj, MS(row, j));
            }
        }
        __syncthreads();
    }

    float* Cb = Cinv + (size_t)b * R_ * R_;
    for (int j = j0; j < j0 + 64; ++j)
        Cb[row * R_ + j] = g2 * MS(row, j);
#undef MS
}

// ---------------------------------------------------------------------------
// Kernel 3: P = (2g*Cinv) @ N -> [128 x 4096] per batch (row-major).
// ---------------------------------------------------------------------------
__global__ __launch_bounds__(32) void k_p(const float* __restrict__ Cinv,
                                          const float* __restrict__ Nmat,
                                          float* __restrict__ P) {
    const int ns = blockIdx.x;
    const int mt = blockIdx.y;
    const int b  = blockIdx.z;
    const int lane = threadIdx.x;
    const int half = lane >> 4;
    const int lm   = lane & 15;

    const float* Cb = Cinv + (size_t)b * R_ * R_;
    const float* Nb = Nmat + (size_t)b * R_ * W_;
    float* Pb = P + (size_t)b * R_ * W_;

    const int wbase = ns * 256;
    const float* arow = Cb + (size_t)(mt * 16 + lm) * R_;

    v8f z = {0.f, 0.f, 0.f, 0.f, 0.f, 0.f, 0.f, 0.f};
    v8f acc[16];
#pragma unroll
    for (int j = 0; j < 16; ++j) acc[j] = z;

    for (int k = 0; k < R_; k += 4) {
        const int kr = k + 2 * half;
        v2f a = *(const v2f*)(arow + kr);
#pragma unroll
        for (int j = 0; j < 16; ++j) {
            const int col = wbase + j * 16 + lm;
            v2f bb;
            bb.x = Nb[(size_t)kr * W_ + col];
            bb.y = Nb[(size_t)(kr + 1) * W_ + col];
            acc[j] = wmma_f32(a, bb, acc[j]);
        }
    }

#pragma unroll
    for (int j = 0; j < 16; ++j)
#pragma unroll
        for (int g = 0; g < 8; ++g) {
            const int row = mt * 16 + g + 8 * half;
            const int col = wbase + j * 16 + lm;
            Pb[(size_t)row * W_ + col] = acc[j][g];
        }
}

// ---------------------------------------------------------------------------
// Kernel 4: out = Ak_L @ P^T per batch. HBM streamer over Ak_L (non-temporal).
// 256 threads = 8 waves; each wave owns 32 rows (2 M-tiles x 8 N-tiles).
// P is staged through double-buffered LDS K-slices by the Tensor Data Mover
// (one tensor_load_to_lds per slice, TENSORcnt + barrier sync), so each block
// reads P exactly once: L2 traffic for B drops from ~4.3 GB to 256 MB total.
// LDS row stride 132 floats -> ds_load_b64 B-fragments are bank-conflict-free.
// ---------------------------------------------------------------------------
__global__ __launch_bounds__(256) void k_x(const float* __restrict__ A,
                                           const float* __restrict__ P,
                                           float* __restrict__ X) {
    extern __shared__ float sm[];
    const int rb   = blockIdx.x;          // 0..15 : 256-row block of H
    const int b    = blockIdx.y;
    const int tid  = threadIdx.x;
    const int wave = tid >> 5;
    const int lane = tid & 31;
    const int half = lane >> 4;
    const int lm   = lane & 15;

    const float* Ab = A + (size_t)b * H_ * W_;
    const float* Pb = P + (size_t)b * R_ * W_;
    float* Xb = X + (size_t)b * H_ * R_;

    const int row0 = rb * 256 + wave * 32;
    const float* arow0 = Ab + (size_t)(row0 + lm) * W_;
    const float* arow1 = Ab + (size_t)(row0 + 16 + lm) * W_;

    v8f z = {0.f, 0.f, 0.f, 0.f, 0.f, 0.f, 0.f, 0.f};
    v8f acc[2][8];
#pragma unroll
    for (int m = 0; m < 2; ++m)
#pragma unroll
        for (int j = 0; j < 8; ++j) acc[m][j] = z;

#if HAVE_TDM
    // Generic LDS pointer low 32 bits == LDS byte offset (ISA flat-LDS rule).
    const unsigned lds0 = (unsigned)(size_t)(void*)sm;
    if (tid < 32) tdm_load_slice(Pb, lds0);           // prefetch slice 0
#endif

    for (int s = 0; s < NSLICE; ++s) {
        float* buf = sm + (size_t)(s & 1) * SLICE_F;
#if HAVE_TDM
        if (tid < 32) {
            if (s + 1 < NSLICE) {
                tdm_load_slice(Pb + (size_t)(s + 1) * KC_,
                               lds0 + (unsigned)(((s + 1) & 1) * SLICE_F * 4));
                __builtin_amdgcn_s_wait_tensorcnt((short)1);  // slice s done
            } else {
                __builtin_amdgcn_s_wait_tensorcnt((short)0);
            }
        }
        __syncthreads();
#else
        __syncthreads();
        for (int i = tid; i < R_ * KC_; i += 256) {
            const int r = i >> 7, c = i & (KC_ - 1);
            buf[r * LROW_ + c] = Pb[(size_t)r * W_ + s * KC_ + c];
        }
        __syncthreads();
#endif
        const int kbase = s * KC_;
        for (int kl = 0; kl < KC_; kl += 4) {
            const int kk = kl + 2 * half;
            v2f a0 = __builtin_nontemporal_load((const v2f*)(arow0 + kbase + kk));
            v2f a1 = __builtin_nontemporal_load((const v2f*)(arow1 + kbase + kk));
#pragma unroll
            for (int j = 0; j < 8; ++j) {
                v2f bb = *(const v2f*)(buf + (j * 16 + lm) * LROW_ + kk);
                acc[0][j] = wmma_f32(a0, bb, acc[0][j]);
                acc[1][j] = wmma_f32(a1, bb, acc[1][j]);
            }
        }
        __syncthreads();   // all reads of buf done before TDM reuses it
    }

#pragma unroll
    for (int m = 0; m < 2; ++m)
#pragma unroll
        for (int j = 0; j < 8; ++j)
#pragma unroll
            for (int g = 0; g < 8; ++g) {
                const int row = row0 + m * 16 + g + 8 * half;
                const int col = j * 16 + lm;
                Xb[(size_t)row * R_ + col] = acc[m][j][g];
            }
}

// ---------------------------------------------------------------------------
extern "C" void kernel_launch(void* const* d_in, const int* in_sizes, int n_in,
                              void* d_out, int out_size, void* d_ws, size_t ws_size,
                              hipStream_t stream) {
    const float* Nmat  = (const float*)d_in[0];   // [B,1,R,W]
    const float* Ak_L  = (const float*)d_in[1];   // [B,1,H,W]
    const float* alpha = (const float*)d_in[2];   // [B]
    const float* gamma = (const float*)d_in[3];   // [B]
    float* out = (float*)d_out;                   // [B,1,H,R]

    char* ws = (char*)d_ws;
    float* G    = (float*)ws;                                    // 512 KB
    float* Cinv = (float*)(ws + (size_t)B_ * R_ * R_ * 4);       // 512 KB
    float* P    = (float*)(ws + (size_t)2 * B_ * R_ * R_ * 4);   // 16 MB

    const int invLds = 128 * 129 * (int)sizeof(float);           // 66,048 B
    const int xLds   = 2 * SLICE_F * (int)sizeof(float);         // 135,168 B

    // Allow >64KB dynamic LDS (immediate host-side calls; capture-safe).
    (void)hipFuncSetAttribute((const void*)k_inv,
                              hipFuncAttributeMaxDynamicSharedMemorySize, invLds);
    (void)hipFuncSetAttribute((const void*)k_x,
                              hipFuncAttributeMaxDynamicSharedMemorySize, xLds);

    k_gram<<<dim3(8, B_), 32, 0, stream>>>(Nmat, G);
    k_inv<<<dim3(B_), 256, invLds, stream>>>(G, alpha, gamma, Cinv);
    k_p<<<dim3(16, 8, B_), 32, 0, stream>>>(Cinv, Nmat, P);
    k_x<<<dim3(16, B_), 256, xLds, stream>>>(Ak_L, P, out);
}